// Loss4RobustPixelReconstruction_OcclusionAware_85203561218656
// MI455X (gfx1250) — compile-verified
//
#include <hip/hip_runtime.h>
#include <math.h>
#include <stdint.h>

// Problem constants (from reference)
#define NB   4
#define CH   3
#define HH   256
#define WW   256
#define KF   11
#define PADR 5
#define EPS2 1e-6f
#define INV_COUNT (1.0f / (4.0f * 3.0f * 256.0f * 256.0f))

// LDS tile: 3 channels x 14 rows x 272 cols
// col offset +9 so that a thread with w0 % 4 == 0 reads 16B-aligned segments
#define TROWS 14
#define TCOLS 272
#define COFF  9
#define TILE_ELEMS (CH * TROWS * TCOLS)

typedef float v2f  __attribute__((ext_vector_type(2)));
typedef float v4f  __attribute__((ext_vector_type(4)));
typedef float v8f  __attribute__((ext_vector_type(8)));

__global__ void ff_zero_out(float* out) {
    out[0] = 0.0f;
}

__global__ __launch_bounds__(256)
void filterflow_charbonnier_kernel(const float* __restrict__ img1,
                                   const float* __restrict__ img2,
                                   const float* __restrict__ filt,
                                   const float* __restrict__ mask,
                                   float* __restrict__ out) {
    __shared__ __align__(16) float tile[CH][TROWS][TCOLS];
    __shared__ float red[16];

    const int tid = threadIdx.x;
    const int b   = blockIdx.x;           // 0..255
    const int n   = b >> 6;               // image index
    const int h0  = (b & 63) << 2;        // first of 4 rows handled by block
    const int r   = tid >> 6;             // row-in-block 0..3
    const int h   = h0 + r;
    const int w0  = (tid & 63) << 2;      // 4 consecutive pixels per thread

    // ---- Stage image1 tile via CDNA5 async-to-LDS (ASYNCcnt path) ----
    // 1) zero the whole tile (covers the padded halo) with vector LDS stores
    {
        v4f z = {0.0f, 0.0f, 0.0f, 0.0f};
        v4f* t4 = (v4f*)&tile[0][0][0];
        for (int q = tid; q < TILE_ELEMS / 4; q += 256) t4[q] = z;
    }
    __syncthreads();  // zeros visible before async writes can land

    // 2) async-load the valid interior: one GLOBAL_LOAD_ASYNC_TO_LDS_B32 per
    //    element; out-of-range rows are EXEC-masked off (per-lane skip)
    for (int idx = tid; idx < CH * TROWS * WW; idx += 256) {
        const int c  = idx / (TROWS * WW);
        const int rm = idx - c * (TROWS * WW);
        const int rr = rm >> 8;        // row in tile
        const int gw = rm & 255;       // global col
        const int gh = h0 - PADR + rr; // global row
        if ((unsigned)gh < (unsigned)HH) {
            const uint32_t goff  = (uint32_t)(((((n * CH) + c) << 16) +
                                              (gh << 8) + gw) * 4);
            // low 32 bits of a generic __shared__ pointer == LDS byte address
            const uint32_t laddr = (uint32_t)(uintptr_t)&tile[c][rr][gw + COFF];
            asm volatile("global_load_async_to_lds_b32 %0, %1, %2"
                         :: "v"(laddr), "v"(goff), "s"(img1) : "memory");
        }
    }
#if __has_builtin(__builtin_amdgcn_s_wait_asynccnt)
    __builtin_amdgcn_s_wait_asynccnt(0);
#else
    asm volatile("s_wait_asynccnt 0" ::: "memory");
#endif
    __syncthreads();

    // ---- Main accumulation: 121 taps, 3 channels, 4 pixels per thread ----
    float acc[CH][4];
#pragma unroll
    for (int c = 0; c < CH; ++c)
#pragma unroll
        for (int p = 0; p < 4; ++p) acc[c][p] = 0.0f;

    // filter base for this (n, h, w0); tap k stride is H*W elements
    const int fbase = ((n * (KF * KF)) << 16) + (h << 8) + w0;
    const int lbase = w0 + 4;  // lds col of (global w0 - 5); multiple of 4

    for (int i = 0; i < KF; ++i) {
        // prefetch next tap-row of filters into the cache hierarchy
        if (i + 1 < KF)
            __builtin_prefetch(&filt[fbase + (((i + 1) * KF) << 16)], 0, 0);

        // load the 3-channel sliding windows (14 live floats, read 16 aligned)
        float seg[CH][16];
#pragma unroll
        for (int c = 0; c < CH; ++c) {
#pragma unroll
            for (int q = 0; q < 4; ++q) {
                const v4f t = *(const v4f*)&tile[c][r + i][lbase + 4 * q];
                seg[c][4 * q + 0] = t[0];
                seg[c][4 * q + 1] = t[1];
                seg[c][4 * q + 2] = t[2];
                seg[c][4 * q + 3] = t[3];
            }
        }

#pragma unroll
        for (int j = 0; j < KF; ++j) {
            // one 16B non-temporal load of filters feeds 12 FMAs
            const v4f f = __builtin_nontemporal_load(
                (const v4f*)&filt[fbase + (((i * KF) + j) << 16)]);
#pragma unroll
            for (int p = 0; p < 4; ++p) {
                const float fp = f[p];
#pragma unroll
                for (int c = 0; c < CH; ++c)
                    acc[c][p] = fmaf(seg[c][j + p], fp, acc[c][p]);
            }
        }
    }

    // ---- Charbonnier loss, masked, pre-scaled ----
    const int pbase = (n << 16) + (h << 8) + w0;
    const v4f m = __builtin_nontemporal_load((const v4f*)&mask[pbase]);
    float loss = 0.0f;
#pragma unroll
    for (int c = 0; c < CH; ++c) {
        const v4f t2 = __builtin_nontemporal_load(
            (const v4f*)&img2[(((n * CH) + c) << 16) + (h << 8) + w0]);
#pragma unroll
        for (int p = 0; p < 4; ++p) {
            const float d = acc[c][p] - t2[p];
            loss += m[p] * sqrtf(d * d + EPS2);
        }
    }
    loss *= INV_COUNT;

    // ---- Wave reduction via V_WMMA_F32_16X16X4_F32 (exact f32 row-sums) ----
    const int lane = tid & 31;
    const int wv   = tid >> 5;
#if __has_builtin(__builtin_amdgcn_wmma_f32_16x16x4_f32)
    // A (16x4 f32): lane L%16 row; K0 (lanes 0-15) / K2 (lanes 16-31) = loss
    v2f a; a[0] = loss; a[1] = 0.0f;
    v2f bo; bo[0] = 1.0f; bo[1] = 1.0f;          // B = all ones (4x16)
    v8f d = {};
    d = __builtin_amdgcn_wmma_f32_16x16x4_f32(false, a, false, bo,
                                              (short)0, d, false, false);
    // D[m][n] = loss[m] + loss[m+16] for every n. Sum 8 rows held per lane:
    float s = d[0] + d[1] + d[2] + d[3] + d[4] + d[5] + d[6] + d[7];
    // lanes 0-15 hold sum of lanes {0..7,16..23}; lanes 16-31 of {8..15,24..31}
    if (lane == 0)  red[wv * 2 + 0] = s;
    if (lane == 16) red[wv * 2 + 1] = s;
#else
    float s = loss;
    for (int off = 16; off > 0; off >>= 1)
        s += __shfl_xor(s, off, 32);
    if (lane == 0) { red[wv * 2 + 0] = s; red[wv * 2 + 1] = 0.0f; }
#endif
    __syncthreads();

    if (tid == 0) {
        float t = 0.0f;
#pragma unroll
        for (int q = 0; q < 16; ++q) t += red[q];
        atomicAdd(out, t);
    }
}

extern "C" void kernel_launch(void* const* d_in, const int* in_sizes, int n_in,
                              void* d_out, int out_size, void* d_ws, size_t ws_size,
                              hipStream_t stream) {
    const float* img1 = (const float*)d_in[0];
    const float* img2 = (const float*)d_in[1];
    const float* filt = (const float*)d_in[2];
    const float* mask = (const float*)d_in[3];
    float* out = (float*)d_out;

    ff_zero_out<<<1, 1, 0, stream>>>(out);
    filterflow_charbonnier_kernel<<<NB * (HH / 4), 256, 0, stream>>>(
        img1, img2, filt, mask, out);
}